// GaussConvLayer_simple_19851338842301
// MI455X (gfx1250) — compile-verified
//
#include <hip/hip_runtime.h>

// ---------------------------------------------------------------------------
// Gaussian "same" convolution, T=16384, nk=16383, via fp32 WMMA on gfx1250.
//
// out[i] = sum_{d=-8191}^{8191} w[d] * xpad[i-d]
//   w[d] = exp(-((d-0.5)-mu)^2 / (2 s^2)) / (s*sqrt(2*pi))
//
// Block-Toeplitz GEMM:  Out[r,m] = sum_e A_e[r,s] * X[s, m-e]
//   A_e[r,s] = w[16e + r - s]   (16x16 Gaussian slab, read from LDS table)
//   X block for shift e = contiguous x[16*(M-e) .. +255]  (16x16, col-major)
// Each e-step = four V_WMMA_F32_16X16X4_F32 (K=16 split into 4).
// ---------------------------------------------------------------------------

typedef __attribute__((ext_vector_type(2))) float v2f;
typedef __attribute__((ext_vector_type(8))) float v8f;

#define T_LEN   16384
#define NCOLS   1024            // T/16 column blocks
#define EMAX    512             // |e|>512 => slab entirely zero
#define WOFF    8208            // table offset: d in [-8207, 8207]
#define WTAB    16416           // padded Gaussian table length (floats)
#define WAVES   8
#define BLOCK   256

__global__ __launch_bounds__(BLOCK)
void gauss_conv_wmma_f32(const float* __restrict__ x,
                         const float* __restrict__ sigma,
                         const float* __restrict__ mu,
                         float* __restrict__ out)
{
    extern __shared__ float smem[];
    float* wlds  = smem;            // [WTAB]  padded Gaussian taps
    float* parts = smem + WTAB;     // [WAVES*256] per-wave partial tiles

    const int tid  = threadIdx.x;
    const int lane = tid & 31;
    const int wave = tid >> 5;
    const int Mt   = blockIdx.x * 16;          // base column block of this tile

    // ---- build padded Gaussian table in LDS (zero outside |d| <= 8191) ----
    const float s      = sigma[0];
    const float m      = mu[0];
    const float inv2s2 = 1.0f / (2.0f * s * s);
    const float norm   = 0.39894228040143267794f / s;   // 1/(s*sqrt(2*pi))

    for (int t = tid; t < WTAB; t += BLOCK) {
        const int d = t - WOFF;
        float val = 0.0f;
        if (d >= -8191 && d <= 8191) {
            const float df = (float)d - 0.5f - m;
            val = norm * expf(-df * df * inv2s2);
        }
        wlds[t] = val;
    }
    __syncthreads();

    // ---- split the e-range across the 8 waves (static => deterministic) ----
    int elo = Mt - 1023; if (elo < -EMAX) elo = -EMAX;   // x-block must touch [0,T)
    int ehi = Mt + 15;   if (ehi >  EMAX) ehi =  EMAX;
    const int total = ehi - elo + 1;
    const int per   = (total + WAVES - 1) / WAVES;
    const int e0    = elo + wave * per;
    int       e1    = e0 + per - 1; if (e1 > ehi) e1 = ehi;

    const int r  = lane & 15;    // A row  / B column / D column index
    const int hi = lane >> 4;    // selects K pair {0,1} vs {2,3}

    v8f acc = {};                // 16x16 fp32 accumulator tile (8 VGPRs/lane)

    for (int e = e0; e <= e1; ++e) {
        // A taps: lane holds A_e[r, s] for s = 4q + 2*hi + v ; d = 16e + r - s
        const int dbase = 16 * e + r - 2 * hi + WOFF;    // index for v=0 at q=0
        // B block: column (Mt - e + n), n == lane&15 == r ; 16 contiguous floats
        const int col = Mt - e + r;
        const bool ok = (col >= 0) && (col < NCOLS);
        const float* xb = x + col * 16 + 2 * hi;

#pragma unroll
        for (int q = 0; q < 4; ++q) {
            v2f a, b;
            a.x = wlds[dbase - 4 * q];        // v = 0
            a.y = wlds[dbase - 4 * q - 1];    // v = 1
            if (ok) {
                b = *(const v2f*)(xb + 4 * q);   // 8B aligned global load
            } else {
                b = (v2f){0.0f, 0.0f};
            }
            // D = A(16x4) x B(4x16) + C, fp32 in/out
            acc = __builtin_amdgcn_wmma_f32_16x16x4_f32(
                      /*neg_a=*/false, a, /*neg_b=*/false, b,
                      /*c_mod=*/(short)0, acc,
                      /*reuse_a=*/false, /*reuse_b=*/false);
        }
    }

    // ---- stash per-wave partial tiles ----
#pragma unroll
    for (int g = 0; g < 8; ++g)
        parts[wave * 256 + lane * 8 + g] = acc[g];
    __syncthreads();

    // ---- fixed-order reduction across the 8 waves, then store ----
    {
        float sum = 0.0f;
#pragma unroll
        for (int w = 0; w < WAVES; ++w)
            sum += parts[w * 256 + tid];
        const int l  = tid >> 3;          // originating lane
        const int g  = tid & 7;           // originating acc VGPR
        // C/D layout: VGPR g holds row (g + 8*(l>>4)), column (l & 15)
        const int oi = 16 * (Mt + (l & 15)) + g + 8 * (l >> 4);
        out[oi] = sum;
    }
}

extern "C" void kernel_launch(void* const* d_in, const int* in_sizes, int n_in,
                              void* d_out, int out_size, void* d_ws, size_t ws_size,
                              hipStream_t stream) {
    const float* x     = (const float*)d_in[0];
    const float* sigma = (const float*)d_in[1];
    const float* mu    = (const float*)d_in[2];
    float*       out   = (float*)d_out;

    const size_t shmem = (size_t)(WTAB + WAVES * 256) * sizeof(float); // ~72 KB

    // Allow >64KB dynamic LDS (gfx1250 WGP has 320KB). Ignore failure.
    (void)hipFuncSetAttribute((const void*)gauss_conv_wmma_f32,
                              hipFuncAttributeMaxDynamicSharedMemorySize,
                              (int)shmem);

    gauss_conv_wmma_f32<<<dim3(T_LEN / 256), dim3(BLOCK), shmem, stream>>>(
        x, sigma, mu, out);
}